// RegionalAttention_3805341024548
// MI455X (gfx1250) — compile-verified
//
#include <hip/hip_runtime.h>
#include <math.h>

#define B_ 8
#define C_ 256
#define H_ 128
#define W_ 128
#define HW_ (H_*W_)
#define C3_ (3*C_)
#define SCALE_ 0.17677669529663687f   // (256/8)^-0.5

typedef __attribute__((ext_vector_type(2))) float v2f;
typedef __attribute__((ext_vector_type(8))) float v8f;

#if __has_builtin(__builtin_amdgcn_wmma_f32_16x16x4_f32)
#define HAS_WMMA_F32X4 1
#endif

// CDNA5 async global->LDS copies (ASYNCcnt-tracked, bypass VGPRs)
#if __has_builtin(__builtin_amdgcn_global_load_async_to_lds_b32) && \
    __has_builtin(__builtin_amdgcn_global_load_async_to_lds_b128)
#define HAS_ASYNC_LDS 1
#endif

#ifdef HAS_ASYNC_LDS
typedef __attribute__((__vector_size__(16))) int i32x4_t;
typedef __attribute__((address_space(1))) int      gbl_i32;
typedef __attribute__((address_space(3))) int      lds_i32;
typedef __attribute__((address_space(1))) i32x4_t  gbl_i32x4;
typedef __attribute__((address_space(3))) i32x4_t  lds_i32x4;

__device__ __forceinline__ void async_b32(const float* g, float* l) {
  __builtin_amdgcn_global_load_async_to_lds_b32((gbl_i32*)g, (lds_i32*)l, 0, 0);
}
__device__ __forceinline__ void async_b128(const float* g, float* l) {
  __builtin_amdgcn_global_load_async_to_lds_b128((gbl_i32x4*)g, (lds_i32x4*)l, 0, 0);
}
#if __has_builtin(__builtin_amdgcn_s_wait_asynccnt)
#define WAIT_ASYNC() __builtin_amdgcn_s_wait_asynccnt(0)
#else
#define WAIT_ASYNC() asm volatile("s_wait_asynccnt 0" ::: "memory")
#endif
#endif

// D = A(16x4) * B(4x16) + C, fp32 in / fp32 accumulate.
__device__ __forceinline__ v8f wmma4(v2f a, v2f b, v8f c) {
#ifdef HAS_WMMA_F32X4
  return __builtin_amdgcn_wmma_f32_16x16x4_f32(false, a, false, b, (short)0, c,
                                               false, false);
#else
  c[0] += a[0]*b[0]; c[1] += a[1]*b[1];
  return c;
#endif
}

// ---------------- stage 0: regional mean over H*W per (b,c) ----------------
__global__ void __launch_bounds__(256) reduce_mean_kernel(
    const float* __restrict__ x, float* __restrict__ regional) {
  int bc = blockIdx.x;
  const float* p = x + (size_t)bc * HW_;
  float s = 0.f;
  for (int i = threadIdx.x; i < HW_; i += 256) s += p[i];
  for (int m = 16; m >= 1; m >>= 1) s += __shfl_xor(s, m, 32);
  __shared__ float red[8];
  int wave = threadIdx.x >> 5, lane = threadIdx.x & 31;
  if (lane == 0) red[wave] = s;
  __syncthreads();
  if (threadIdx.x == 0) {
    float t = 0.f;
    for (int i = 0; i < 8; i++) t += red[i];
    regional[bc] = t * (1.0f / HW_);
  }
}

// out[b,o] = bias[o] + sum_k wgt[o,k]*in[b,k]
__global__ void __launch_bounds__(256) smallfc_kernel(
    const float* __restrict__ wgt, const float* __restrict__ bias,
    const float* __restrict__ in, float* __restrict__ out) {
  int b = blockIdx.x, o = threadIdx.x;
  const float* wr = wgt + (size_t)o * C_;
  const float* ir = in + (size_t)b * C_;
  float s = bias[o];
  for (int k = 0; k < C_; k++) s = fmaf(wr[k], ir[k], s);
  out[(size_t)b * C_ + o] = s;
}

// ---------------- WMMA GEMM: Dst[b,o,hw] = A[o,:]*X[b,:,hw] + bias ---------
// Block tile 64(M) x 64(N), 8 waves = 4x2, each wave 16x32.
// Double-buffered LDS panels filled by async global->LDS copies.
template <int MROWS, bool QKV_MODE>
__global__ void __launch_bounds__(256) gemm_wmma_kernel(
    const float* __restrict__ A, const float* __restrict__ X,
    const float* __restrict__ bias, float* __restrict__ Dst) {
  __shared__ float As[2][64][20];   // pad 20: 16B-aligned rows + conflict-free
  __shared__ float Bs[2][16][64];
  int p0  = blockIdx.x * 64;
  int o0  = blockIdx.y * 64;
  int b   = p0 >> 14;               // / HW_
  int hw0 = p0 & (HW_ - 1);
  int tid = threadIdx.x;
  int wave = tid >> 5, lane = tid & 31;
  int half = lane >> 4, ln = lane & 15;
  int wm = wave >> 1, wn = wave & 1;

  v8f acc0 = {0.f,0.f,0.f,0.f,0.f,0.f,0.f,0.f};
  v8f acc1 = {0.f,0.f,0.f,0.f,0.f,0.f,0.f,0.f};

  // per-thread staging coordinates: one b128 per array per chunk
  int rA = tid >> 2, k4A = (tid & 3) * 4;     // As: 64 x 16 / 4
  int kB = tid >> 4, j4B = (tid & 15) * 4;    // Bs: 16 x 64 / 4

#ifdef HAS_ASYNC_LDS
  async_b128(&A[(size_t)(o0 + rA) * C_ + k4A], &As[0][rA][k4A]);
  async_b128(&X[((size_t)b * C_ + kB) * HW_ + hw0 + j4B], &Bs[0][kB][j4B]);
  WAIT_ASYNC();
  __syncthreads();
#endif

  for (int chunk = 0; chunk < 16; chunk++) {
    int kc = chunk * 16;
#ifdef HAS_ASYNC_LDS
    int cur = chunk & 1;
    if (chunk + 1 < 16) {   // issue next panel while computing this one
      int nxt = cur ^ 1, kn = kc + 16;
      async_b128(&A[(size_t)(o0 + rA) * C_ + kn + k4A], &As[nxt][rA][k4A]);
      async_b128(&X[((size_t)b * C_ + kn + kB) * HW_ + hw0 + j4B], &Bs[nxt][kB][j4B]);
    }
#else
    const int cur = 0;
    __syncthreads();
    {
      float* dA = &As[0][rA][k4A];
      const float* sA = &A[(size_t)(o0 + rA) * C_ + kc + k4A];
      for (int q = 0; q < 4; q++) dA[q] = sA[q];
      float* dB = &Bs[0][kB][j4B];
      const float* sB = &X[((size_t)b * C_ + kc + kB) * HW_ + hw0 + j4B];
      for (int q = 0; q < 4; q++) dB[q] = sB[q];
    }
    __syncthreads();
#endif
#pragma unroll
    for (int kk = 0; kk < 4; kk++) {
      int kb = kk * 4 + 2 * half;   // frag K index per ISA layout
      v2f av, bv0, bv1;
      av[0]  = As[cur][wm * 16 + ln][kb + 0];
      av[1]  = As[cur][wm * 16 + ln][kb + 1];
      bv0[0] = Bs[cur][kb + 0][wn * 32 + ln];
      bv0[1] = Bs[cur][kb + 1][wn * 32 + ln];
      bv1[0] = Bs[cur][kb + 0][wn * 32 + 16 + ln];
      bv1[1] = Bs[cur][kb + 1][wn * 32 + 16 + ln];
      acc0 = wmma4(av, bv0, acc0);
      acc1 = wmma4(av, bv1, acc1);
    }
#ifdef HAS_ASYNC_LDS
    WAIT_ASYNC();
    __syncthreads();
#endif
  }
#pragma unroll
  for (int r = 0; r < 8; r++) {
    int o = o0 + wm * 16 + r + 8 * half;
    float bb, sc = 1.f;
    if (QKV_MODE) { bb = bias[o]; if (o < C_) sc = SCALE_; }  // scale Q rows
    else          { bb = bias[(size_t)b * C_ + o]; }          // pbias[b,o]
    size_t base = ((size_t)b * MROWS + o) * HW_ + hw0;
    Dst[base + wn * 32 + ln]      = (acc0[r] + bb) * sc;
    Dst[base + wn * 32 + 16 + ln] = (acc1[r] + bb) * sc;
  }
}

// ---------------- flash-style axial attention over H, one block per (b,w) --
// Dynamic LDS (~210 KB of the 320 KB/WGP): full Q panel + double-buffered K/V.
struct AttnLds {
  float Qs[C_][H_];       // [c][i]     128 KB
  float Ks[2][C_][17];    // [buf][c][j] 34 KB
  float Vs[2][C_][17];    //             34 KB
  float Ps[8][16][17];    // per-wave P tile (17-pad => conflict-free T reads)
  float Row[8][16];       // per-wave per-row broadcast scratch
};

__global__ void __launch_bounds__(256, 1) attn_kernel(
    const float* __restrict__ qkv, float* __restrict__ att) {
  extern __shared__ char smem_raw[];
  AttnLds& L = *reinterpret_cast<AttnLds*>(smem_raw);

  int blk = blockIdx.x;
  int b = blk >> 7, w = blk & (W_ - 1);
  int tid = threadIdx.x;
  int wave = tid >> 5, lane = tid & 31;
  int half = lane >> 4, ln = lane & 15;
  int i0 = wave * 16;
  int irow = i0 + ln;

  const float* qbase = qkv + ((size_t)b * C3_         ) * HW_ + w;
  const float* kbase = qkv + ((size_t)b * C3_ +     C_) * HW_ + w;
  const float* vbase = qkv + ((size_t)b * C3_ + 2 * C_) * HW_ + w;

  // prologue: stage whole Q panel + first K/V panel
#ifdef HAS_ASYNC_LDS
  for (int idx = tid; idx < C_ * H_; idx += 256) {
    int c = idx >> 7, i = idx & 127;
    async_b32(&qbase[(size_t)c * HW_ + i * W_], &L.Qs[c][i]);
  }
  for (int idx = tid; idx < C_ * 16; idx += 256) {
    int c = idx >> 4, j = idx & 15;
    async_b32(&kbase[(size_t)c * HW_ + j * W_], &L.Ks[0][c][j]);
    async_b32(&vbase[(size_t)c * HW_ + j * W_], &L.Vs[0][c][j]);
  }
  WAIT_ASYNC();
#else
  for (int idx = tid; idx < C_ * H_; idx += 256) {
    int c = idx >> 7, i = idx & 127;
    L.Qs[c][i] = qbase[(size_t)c * HW_ + i * W_];
  }
#endif
  __syncthreads();

  v8f oacc[16];   // O tiles: 16 c-tiles x (16c x 16i)
#pragma unroll
  for (int ct = 0; ct < 16; ct++)
    oacc[ct] = (v8f){0.f,0.f,0.f,0.f,0.f,0.f,0.f,0.f};
  float runM[8], runL[8];
#pragma unroll
  for (int r = 0; r < 8; r++) { runM[r] = -3.0e38f; runL[r] = 0.f; }

  for (int jb = 0; jb < 8; jb++) {
#ifdef HAS_ASYNC_LDS
    int cur = jb & 1;
    if (jb + 1 < 8) {   // prefetch next j-panel while computing this one
      int nxt = cur ^ 1, j0n = (jb + 1) * 16;
      for (int idx = tid; idx < C_ * 16; idx += 256) {
        int c = idx >> 4, j = idx & 15;
        async_b32(&kbase[(size_t)c * HW_ + (j0n + j) * W_], &L.Ks[nxt][c][j]);
        async_b32(&vbase[(size_t)c * HW_ + (j0n + j) * W_], &L.Vs[nxt][c][j]);
      }
    }
#else
    const int cur = 0;
    int j0 = jb * 16;
    __syncthreads();
    for (int idx = tid; idx < C_ * 16; idx += 256) {
      int c = idx >> 4, j = idx & 15;
      L.Ks[0][c][j] = kbase[(size_t)c * HW_ + (j0 + j) * W_];
      L.Vs[0][c][j] = vbase[(size_t)c * HW_ + (j0 + j) * W_];
    }
    __syncthreads();
#endif

    // S tile (16i x 16j) = Q^T K over K=256 (64 wmma steps).
    // Partial unroll: keeps 8 in-flight address pairs instead of 64,
    // holding the wave under 256 VGPRs (avoids s_set_vgpr_msb churn).
    v8f s = {0.f,0.f,0.f,0.f,0.f,0.f,0.f,0.f};
#pragma unroll 8
    for (int c4 = 0; c4 < 64; c4++) {
      int c = c4 * 4 + 2 * half;
      v2f av, kv;
      av[0] = L.Qs[c    ][irow];
      av[1] = L.Qs[c + 1][irow];
      kv[0] = L.Ks[cur][c    ][ln];
      kv[1] = L.Ks[cur][c + 1][ln];
      s = wmma4(av, kv, s);
    }

    // online softmax; row m = r + 8*half within the wave
#pragma unroll
    for (int r = 0; r < 8; r++) {
      float rm = s[r];
      for (int m = 8; m >= 1; m >>= 1) rm = fmaxf(rm, __shfl_xor(rm, m, 32));
      float nm   = fmaxf(runM[r], rm);
      float corr = __expf(runM[r] - nm);
      float pv   = __expf(s[r] - nm);
      float rs   = pv;
      for (int m = 8; m >= 1; m >>= 1) rs += __shfl_xor(rs, m, 32);
      runL[r] = runL[r] * corr + rs;
      runM[r] = nm;
      L.Ps[wave][r + 8 * half][ln] = pv;
      if (ln == 0) L.Row[wave][r + 8 * half] = corr;
    }
    float corrn = L.Row[wave][ln];   // O-tile row index i = N = ln
#pragma unroll
    for (int ct = 0; ct < 16; ct++)
#pragma unroll
      for (int r = 0; r < 8; r++) oacc[ct][r] *= corrn;

    // O[c,i] += sum_j V[c,j] * P[i,j]; bounded unroll for address pressure
#pragma unroll 2
    for (int ct = 0; ct < 16; ct++) {
      int crow = ct * 16 + ln;
#pragma unroll
      for (int kk = 0; kk < 4; kk++) {
        int jloc = kk * 4 + 2 * half;
        v2f av, bv;
        av[0] = L.Vs[cur][crow][jloc + 0];
        av[1] = L.Vs[cur][crow][jloc + 1];
        bv[0] = L.Ps[wave][ln][jloc + 0];
        bv[1] = L.Ps[wave][ln][jloc + 1];
        oacc[ct] = wmma4(av, bv, oacc[ct]);
      }
    }
#ifdef HAS_ASYNC_LDS
    WAIT_ASYNC();
    __syncthreads();
#endif
  }

  // finalize: divide by softmax sums, store [B,C,H,W]
  if (ln == 0) {
#pragma unroll
    for (int r = 0; r < 8; r++) L.Row[wave][r + 8 * half] = 1.0f / runL[r];
  }
  float invl = L.Row[wave][ln];
  int iw = i0 + ln;
#pragma unroll 2
  for (int ct = 0; ct < 16; ct++) {
#pragma unroll
    for (int r = 0; r < 8; r++) {
      int c = ct * 16 + r + 8 * half;
      att[((size_t)b * C_ + c) * HW_ + iw * W_ + w] = oacc[ct][r] * invl;
    }
  }
}

extern "C" void kernel_launch(void* const* d_in, const int* in_sizes, int n_in,
                              void* d_out, int out_size, void* d_ws, size_t ws_size,
                              hipStream_t stream) {
  (void)in_sizes; (void)n_in; (void)out_size; (void)ws_size;
  const float* x      = (const float*)d_in[0];
  const float* qkv_w  = (const float*)d_in[1];
  const float* qkv_b  = (const float*)d_in[2];
  const float* fc_w   = (const float*)d_in[3];
  const float* fc_b   = (const float*)d_in[4];
  const float* proj_w = (const float*)d_in[5];
  const float* proj_b = (const float*)d_in[6];
  float* out = (float*)d_out;

  float* ws       = (float*)d_ws;
  float* qkvbuf   = ws;                                  // 8*768*16384 f
  float* attbuf   = qkvbuf + (size_t)B_ * C3_ * HW_;     // 8*256*16384 f
  float* regional = attbuf + (size_t)B_ * C_ * HW_;      // 2048 f
  float* reg2     = regional + B_ * C_;                  // 2048 f
  float* pbias    = reg2 + B_ * C_;                      // 2048 f

  // regional branch folded into a per-(b,o) proj bias
  reduce_mean_kernel<<<B_ * C_, 256, 0, stream>>>(x, regional);
  smallfc_kernel<<<B_, C_, 0, stream>>>(fc_w, fc_b, regional, reg2);
  smallfc_kernel<<<B_, C_, 0, stream>>>(proj_w, proj_b, reg2, pbias);

  dim3 gq(B_ * HW_ / 64, C3_ / 64);   // 2048 x 12
  gemm_wmma_kernel<C3_, true><<<gq, 256, 0, stream>>>(qkv_w, x, qkv_b, qkvbuf);

  size_t attn_lds = sizeof(AttnLds);  // ~210 KB of the 320 KB/WGP
  (void)hipFuncSetAttribute((const void*)attn_kernel,
                            hipFuncAttributeMaxDynamicSharedMemorySize,
                            (int)attn_lds);
  attn_kernel<<<B_ * W_, 256, attn_lds, stream>>>(qkvbuf, attbuf);

  dim3 gp(B_ * HW_ / 64, C_ / 64);    // 2048 x 4
  gemm_wmma_kernel<C_, false><<<gp, 256, 0, stream>>>(proj_w, attbuf, pbias, out);
}